// ROI_pool_62723702391389
// MI455X (gfx1250) — compile-verified
//
#include <hip/hip_runtime.h>
#include <float.h>

// Problem constants (from reference setup_inputs)
#define OUTB 7
#define CHAN 256
#define HH 50
#define WW 50
#define NROI 128

typedef unsigned int v4u __attribute__((ext_vector_type(4)));
typedef int          v8i __attribute__((ext_vector_type(8)));
typedef int          v4i __attribute__((ext_vector_type(4)));

#define HAVE_TDM (__has_builtin(__builtin_amdgcn_tensor_load_to_lds) && \
                  __has_builtin(__builtin_amdgcn_s_wait_tensorcnt))

// One wave32 block per (roi, channel).
// Stage the ROI sub-tile (h x w, row stride 50) of the feature plane into LDS
// with one Tensor-Data-Mover DMA, wait on TENSORcnt, then do separable
// row-bin / col-bin max pooling from LDS.
__global__ __launch_bounds__(32) void roi_pool_62723702391389_kernel(
    const float* __restrict__ features,   // (B, C, H, W)
    const float* __restrict__ cat_rois,   // (N, 5): im, x1, y1, x2, y2
    float* __restrict__ out)              // (N, C, 7, 7)
{
    __shared__ float tile[HH * WW];        // staged ROI tile (compact h x w)
    __shared__ float rowpool[OUTB * WW];   // 7 row-bin maxima per column

    const int blk = blockIdx.x;
    const int n = blk / CHAN;
    const int c = blk % CHAN;

    // ROI decode: scale by 1/16, round-half-to-even (matches jnp.round).
    const float* r = cat_rois + n * 5;
    const int im = (int)__builtin_rintf(r[0]);
    const int x1 = (int)__builtin_rintf(r[1] * 0.0625f);
    const int y1 = (int)__builtin_rintf(r[2] * 0.0625f);
    const int x2 = (int)__builtin_rintf(r[3] * 0.0625f);
    const int y2 = (int)__builtin_rintf(r[4] * 0.0625f);
    const int h = y2 - y1 + 1;             // 1..50
    const int w = x2 - x1 + 1;             // 1..50

    const float* plane    = features + (size_t)(im * CHAN + c) * (HH * WW);
    const float* tile_src = plane + y1 * WW + x1;   // tile start in memory

    const int t = threadIdx.x;

#if HAVE_TDM
    {
        // LDS byte offset of destination tile (generic LDS ptr low 32 bits).
        const unsigned lds_off = (unsigned)(unsigned long long)(void*)&tile[0];
        const unsigned long long ga = (unsigned long long)tile_src;

        // D# group 0: count=1 | lds_addr | global_addr[56:0] | type=2
        v4u g0 = { 1u,
                   lds_off,
                   (unsigned)(ga & 0xffffffffu),
                   (unsigned)((ga >> 32) & 0x01ffffffu) | (2u << 30) };
        // D# group 1 (dword k = bits [32k+31:32k]):
        //  e0: wg_mask=0, data_size=2 (4B)
        //  e1[31:16] = tensor_dim0 lo (=50)
        //  e2[31:16] = tensor_dim1 lo (=50)
        //  e3[31:16] = tile_dim0 (=w)
        //  e4[15:0]  = tile_dim1 (=h), tile_dim2=0
        //  e5        = tensor_dim0_stride lo (=50)
        v8i g1 = { (int)(2u << 16),
                   (int)(50u << 16),
                   (int)(50u << 16),
                   (int)((unsigned)w << 16),
                   (int)(unsigned)h,
                   50,
                   0, 0 };
        v4i gz = { 0, 0, 0, 0 };
#if __clang_major__ >= 23
        v8i gz8 = { 0, 0, 0, 0, 0, 0, 0, 0 };
        __builtin_amdgcn_tensor_load_to_lds(g0, g1, gz, gz, gz8, 0);
#else
        __builtin_amdgcn_tensor_load_to_lds(g0, g1, gz, gz, 0);
#endif
        __builtin_amdgcn_s_wait_tensorcnt(0);
    }
#else
    // Fallback: cooperative staged copy global -> LDS.
    for (int idx = t; idx < h * w; idx += 32) {
        const int rr = idx / w;
        const int cc = idx - rr * w;
        if (idx + 32 < h * w)
            __builtin_prefetch(tile_src + ((idx + 32) / w) * WW + ((idx + 32) % w), 0, 0);
        tile[idx] = tile_src[rr * WW + cc];
    }
#endif
    __syncthreads();

    // ---- Phase 1: row pooling per column (tile-relative coords). ----
    for (int ct = t; ct < w; ct += 32) {
#pragma unroll
        for (int i = 0; i < OUTB; ++i) {
            const int ys = (i * h) / OUTB;                       // floor
            const int ye = ((i + 1) * h + OUTB - 1) / OUTB;      // ceil <= h
            float m = -FLT_MAX;
            for (int rr = ys; rr < ye; ++rr)
                m = fmaxf(m, tile[rr * w + ct]);
            rowpool[i * WW + ct] = m;
        }
    }
    __syncthreads();

    // ---- Phase 2: column pooling -> 7x7 outputs. ----
    float* oplane = out + (size_t)(n * CHAN + c) * (OUTB * OUTB);
    for (int bt = t; bt < OUTB * OUTB; bt += 32) {
        const int i = bt / OUTB;
        const int j = bt % OUTB;
        const int xs = (j * w) / OUTB;                           // floor
        const int xe = ((j + 1) * w + OUTB - 1) / OUTB;          // ceil <= w
        float m = -FLT_MAX;
        for (int col = xs; col < xe; ++col)
            m = fmaxf(m, rowpool[i * WW + col]);
        oplane[bt] = m;
    }
}

extern "C" void kernel_launch(void* const* d_in, const int* in_sizes, int n_in,
                              void* d_out, int out_size, void* d_ws, size_t ws_size,
                              hipStream_t stream) {
    (void)in_sizes; (void)n_in; (void)out_size; (void)d_ws; (void)ws_size;
    const float* features = (const float*)d_in[0];   // 2*256*50*50 f32
    const float* cat_rois = (const float*)d_in[1];   // 128*5 f32
    float* out = (float*)d_out;                      // 128*256*7*7 f32

    dim3 grid(NROI * CHAN);   // 32768 single-wave blocks
    dim3 block(32);           // 1 x wave32 (TDM issued once per block)
    roi_pool_62723702391389_kernel<<<grid, block, 0, stream>>>(features, cat_rois, out);
}